// JeffressLinear_53910429499967
// MI455X (gfx1250) — compile-verified
//
#include <hip/hip_runtime.h>

// JeffressLinear for MI455X (gfx1250), round 2.
// out[:, d] = (w*L) @ S  per (n,c) block, where
//   L[t,s] = 0.5^(t-s) for s<=t  (shared lower-triangular decay kernel),
//   S[s,d] = x[(s-rd0_d) mod T, j=0] + x[(s-rd1_d) mod T, j=1].
// w*L is precomputed once per block into LDS (A fragments = one ds_load_b64),
// x is async-staged (global_load_async_to_lds_b32) deinterleaved + replicated
// so B gathers are two adjacent-dword LDS reads with incremental addresses.
// Compute: V_WMMA_F32_16X16X4_F32, 8 accumulators (2 N-tiles x 4 M-tiles)
// sharing each A fragment.

typedef __attribute__((ext_vector_type(2))) float v2f;
typedef __attribute__((ext_vector_type(8))) float v8f;

#define T_DIM 64
#define N_DIM 16
#define C_DIM 128
#define D_OUT 256
#define OUT_T_STRIDE ((size_t)(N_DIM) * C_DIM * D_OUT)  // 524288

#if defined(__has_builtin)
#if __has_builtin(__builtin_amdgcn_wmma_f32_16x16x4_f32)
#define HAVE_WMMA_F32X4 1
#endif
#endif

// Generic pointer to a __shared__ object -> raw 32-bit LDS byte offset
// (what GLOBAL_LOAD_ASYNC_TO_LDS expects in its VDST operand; flat LDS
// addresses carry the wave-relative offset in the low 32 bits, ISA 10.2).
__device__ __forceinline__ unsigned lds_offset(const void* p) {
    return (unsigned)(unsigned long long)
        (__attribute__((address_space(3))) const void*)p;
}

__device__ __forceinline__ void async_load_b32_to_lds(unsigned lds_off,
                                                      const void* gaddr) {
    asm volatile("global_load_async_to_lds_b32 %0, %1, off"
                 :: "v"(lds_off), "v"(gaddr)
                 : "memory");
}

__global__ __launch_bounds__(256) void jeffress_wmma2_kernel(
    const float* __restrict__ x,          // (T, N, C, 2)
    const float* __restrict__ log_delay,  // (D_OUT, 1)
    const float* __restrict__ log_weight, // scalar
    const float* __restrict__ noise,      // (N, C, D_OUT, 2)
    float* __restrict__ out)              // (T, N, C, D_OUT)
{
    __shared__ float Lw[T_DIM * T_DIM];   // 16 KB: w-scaled decay kernel [m][k]
    __shared__ float xL0[2 * T_DIM];      // j=0 series, replicated (wrap-free +1)
    __shared__ float xL1[2 * T_DIM];      // j=1 series, replicated
    __shared__ int   rds[2 * D_OUT];      // per-column delays (j=0, then j=1)
    __shared__ int   maxd_s[2];

    const int b   = blockIdx.x;           // n*C + c
    const int tid = threadIdx.x;

    const float w = expf(log_weight[0]);

    // ---- Phase 0: build L' = w * L (once per block; 16 entries/thread) ----
    {
        const int m  = tid & 63;
        const int kb = (tid >> 6) << 4;
        #pragma unroll
        for (int kk = 0; kk < 16; ++kk) {
            const int k = kb + kk;
            Lw[m * T_DIM + k] = (k <= m) ? ldexpf(w, k - m) : 0.0f;
        }
    }

    // ---- Phase 1: async-stage x[:, n, c, :] deinterleaved + replicated ----
    if (tid < 2 * T_DIM) {
        const int t = tid >> 1, j = tid & 1;
        const float* g = x + (size_t)t * (N_DIM * C_DIM * 2) + (size_t)b * 2 + j;
        float* dst = j ? xL1 : xL0;
        async_load_b32_to_lds(lds_offset(&dst[t]), g);
        async_load_b32_to_lds(lds_offset(&dst[t + T_DIM]), g);
    }
    asm volatile("s_wait_asynccnt 0" ::: "memory");
    __syncthreads();

    // ---- Phase 2: first-occurrence argmax over t (j=0,1) ----
    if (tid == 0) {
        float m0 = xL0[0], m1 = xL1[0];
        int a0 = 0, a1 = 0;
        for (int t = 1; t < T_DIM; ++t) {
            const float v0 = xL0[t], v1 = xL1[t];
            if (v0 > m0) { m0 = v0; a0 = t; }
            if (v1 > m1) { m1 = v1; a1 = t; }
        }
        maxd_s[0] = (T_DIM - 1) - a0;
        maxd_s[1] = (T_DIM - 1) - a1;
    }
    __syncthreads();

    // ---- Phase 3: per-column stochastic-rounded, clamped delays ----
    {
        const int d = tid;
        const float del0 = expf(log_delay[d]);
        const float del1 = expf(log_delay[(D_OUT - 1) - d]);  // flipped column
        const float s0 = (float)T_DIM * del0;
        const float s1 = (float)T_DIM * del1;
        const float f0 = floorf(s0), f1 = floorf(s1);
        const float n0 = noise[((size_t)b * D_OUT + d) * 2 + 0];
        const float n1 = noise[((size_t)b * D_OUT + d) * 2 + 1];
        int r0 = (int)f0 + ((n0 < (s0 - f0)) ? 1 : 0);
        int r1 = (int)f1 + ((n1 < (s1 - f1)) ? 1 : 0);
        const int m0 = maxd_s[0], m1 = maxd_s[1];
        rds[d]         = (r0 < m0) ? r0 : m0;
        rds[D_OUT + d] = (r1 < m1) ? r1 : m1;
    }
    __syncthreads();

#if HAVE_WMMA_F32X4
    // ---- Phase 4: Y = L' @ S via V_WMMA_F32_16X16X4_F32 ----
    // Wave layout: wave owns N-tiles {2*wid, 2*wid+1}; A shared across both.
    const int lane = tid & 31;
    const int wid  = tid >> 5;
    const int lrow = lane & 15;          // M (A/C/D) and N (B/C/D) lane index
    const int hi   = lane >> 4;          // lane-half -> K offset 2*hi

    const int colA = wid * 32 + lrow;
    const int colB = colA + 16;
    const int rdA0 = rds[colA], rdA1 = rds[D_OUT + colA];
    const int rdB0 = rds[colB], rdB1 = rds[D_OUT + colB];

    // Gather indices into replicated series: idx(kc) = (4kc + 2hi - rd) & 63.
    int aA0 = ((hi << 1) - rdA0) & 63;
    int aA1 = ((hi << 1) - rdA1) & 63;
    int aB0 = ((hi << 1) - rdB0) & 63;
    int aB1 = ((hi << 1) - rdB1) & 63;

    v8f acc[2][4];
    #pragma unroll
    for (int q = 0; q < 2; ++q)
        #pragma unroll
        for (int mt = 0; mt < 4; ++mt)
            acc[q][mt] = (v8f){0, 0, 0, 0, 0, 0, 0, 0};

    // A row base for this lane: Lw[(mt*16 + lrow)][4*kc + 2*hi]; 8B-aligned.
    const float* Lrow = &Lw[lrow * T_DIM + (hi << 1)];

    #pragma unroll
    for (int kc = 0; kc < 16; ++kc) {
        // B fragments: S[k][col] for k = 4kc + 2hi + {0,1} (adjacent dwords).
        v2f bA = { xL0[aA0] + xL1[aA1], xL0[aA0 + 1] + xL1[aA1 + 1] };
        v2f bB = { xL0[aB0] + xL1[aB1], xL0[aB0 + 1] + xL1[aB1 + 1] };
        #pragma unroll
        for (int mt = 0; mt < 4; ++mt) {
            const v2f a = *(const v2f*)&Lrow[mt * 16 * T_DIM + kc * 4];
            acc[0][mt] = __builtin_amdgcn_wmma_f32_16x16x4_f32(
                false, a, false, bA, (short)0, acc[0][mt], false, false);
            acc[1][mt] = __builtin_amdgcn_wmma_f32_16x16x4_f32(
                false, a, false, bB, (short)0, acc[1][mt], false, false);
        }
        aA0 = (aA0 + 4) & 63;  aA1 = (aA1 + 4) & 63;
        aB0 = (aB0 + 4) & 63;  aB1 = (aB1 + 4) & 63;
    }

    // D layout: VGPR r -> t = mt*16 + r + 8*hi, col = tile_base + (lane&15).
    // Weight already folded into L', so stores are raw accumulator values.
    const size_t obase = (size_t)b * D_OUT;
    const int thi = hi << 3;
    #pragma unroll
    for (int q = 0; q < 2; ++q) {
        const int col = (q == 0) ? colA : colB;
        #pragma unroll
        for (int mt = 0; mt < 4; ++mt) {
            #pragma unroll
            for (int r = 0; r < 8; ++r) {
                const int t = mt * 16 + r + thi;
                out[(size_t)t * OUT_T_STRIDE + obase + col] = acc[q][mt][r];
            }
        }
    }
#else
    // Fallback: sequential leaky-integrator scan (no wmma builtin available).
    {
        const int d   = tid;
        const int rd0 = rds[d];
        const int rd1 = rds[D_OUT + d];
        float y0 = 0.0f, y1 = 0.0f;
        for (int t = 0; t < T_DIM; ++t) {
            y0 = 0.5f * y0 + xL0[(t - rd0) & 63];
            y1 = 0.5f * y1 + xL1[(t - rd1) & 63];
            out[(size_t)t * OUT_T_STRIDE + (size_t)b * D_OUT + d] = w * (y0 + y1);
        }
    }
#endif
}

extern "C" void kernel_launch(void* const* d_in, const int* in_sizes, int n_in,
                              void* d_out, int out_size, void* d_ws, size_t ws_size,
                              hipStream_t stream) {
    (void)in_sizes; (void)n_in; (void)out_size; (void)d_ws; (void)ws_size;
    const float* x          = (const float*)d_in[0];
    const float* log_delay  = (const float*)d_in[1];
    const float* log_weight = (const float*)d_in[2];
    const float* noise      = (const float*)d_in[3];
    float* out              = (float*)d_out;

    const dim3 grid(N_DIM * C_DIM);   // 2048 blocks, one per (n,c)
    const dim3 block(256);            // 8 waves (wave32)
    jeffress_wmma2_kernel<<<grid, block, 0, stream>>>(x, log_delay, log_weight,
                                                      noise, out);
}